// EdgeAwareLoss_52278341927513
// MI455X (gfx1250) — compile-verified
//
#include <hip/hip_runtime.h>
#include <hip/hip_bf16.h>
#include <math.h>

#define HH 1024
#define WW 1024
#define TG22f 0.4142135623730951f
#define TG67f 2.414213562373095f
#define LOWT 50.0f
#define HIGHT 150.0f
#define EDGEW 5.0f

typedef __attribute__((ext_vector_type(2))) float v2f;
typedef __attribute__((ext_vector_type(8))) float v8f;
typedef __attribute__((ext_vector_type(4))) unsigned int u32x4;
typedef __attribute__((ext_vector_type(8))) unsigned int u32x8;

// Wave32 sum using V_WMMA_F32_16X16X4_F32 with B = ones:
// A[16x4] holds lane partials, D = A*ones -> row sums replicated across N.
// Lane j (0-15) sums c[0..7] = rows 0..7; lane j+16 sums rows 8..15; xor-16 combine.
__device__ __forceinline__ float wave_sum_wmma(float p) {
    v2f a; a[0] = p;    a[1] = 0.0f;
    v2f b; b[0] = 1.0f; b[1] = 1.0f;
    v8f c = {};
    c = __builtin_amdgcn_wmma_f32_16x16x4_f32(false, a, false, b, (short)0, c, false, false);
    float s = c[0] + c[1] + c[2] + c[3] + c[4] + c[5] + c[6] + c[7];
    s += __shfl_xor(s, 16, 32);
    return s; // full wave sum in every lane
}

// Issue one TDM 2D tile load (fp32 elements) via the Tensor Data Mover.
// D# per CDNA5 ISA ch.8: group0 = {count, lds_addr, global_addr, type=2},
// group1 = {data_size, tensor dims, tile dims, dim0 stride}. Tracked on TENSORcnt.
__device__ __forceinline__ void tdm_load_tile_f32(unsigned long long gaddr, unsigned lds_off,
                                                  unsigned tile_w, unsigned tile_h,
                                                  unsigned tensor_w, unsigned tensor_h,
                                                  unsigned row_stride_elems) {
    u32x4 g0;
    g0[0] = 1u;                                               // count=1, user descriptor
    g0[1] = lds_off;                                          // LDS byte address
    g0[2] = (unsigned)gaddr;                                  // global_addr[31:0]
    g0[3] = (unsigned)((gaddr >> 32) & 0x01FFFFFFu)           // global_addr[56:32]
            | (2u << 30);                                     // type = 2 ("image")
    u32x8 g1;
    g1[0] = (2u << 16);                                       // wg_mask=0, data_size=2 (4B)
    g1[1] = (tensor_w & 0xFFFFu) << 16;                       // tensor_dim0[15:0]
    g1[2] = (tensor_w >> 16) | ((tensor_h & 0xFFFFu) << 16);  // dim0[31:16] | dim1[15:0]
    g1[3] = (tensor_h >> 16) | (tile_w << 16);                // dim1[31:16] | tile_dim0
    g1[4] = tile_h;                                           // tile_dim1 | tile_dim2=0
    g1[5] = row_stride_elems;                                 // tensor_dim0_stride[31:0]
    g1[6] = 0u;                                               // stride0[47:32] | stride1[15:0]
    g1[7] = 0u;                                               // stride1 unused (2D tile)
    asm volatile("tensor_load_to_lds %0, %1" : : "s"(g0), "s"(g1) : "memory");
}

// ---------------- Kernel 1: binarize + Sobel + NMS -> weak/strong flag byte ----------------
// flags bit0 = cand (weak), bit1 = strong
__global__ void sobel_nms_kernel(const float* __restrict__ target,
                                 unsigned char* __restrict__ flags) {
    __shared__ float img[20][20]; // tile + halo2, edge-clamped (conv pads with 'edge')
    __shared__ float sm[18][18];  // |gx|+|gy|, tile + halo1, zero outside image (mp zero-pad)
    const int tx = threadIdx.x, ty = threadIdx.y;
    const int tid = ty * 16 + tx;
    const int tile_r = blockIdx.y * 16, tile_c = blockIdx.x * 16;
    const int b = blockIdx.z;
    const float* tb = target + (size_t)b * HH * WW;

    for (int i = tid; i < 400; i += 256) {
        int r = i / 20, c = i % 20;
        int gr = tile_r + r - 2, gc = tile_c + c - 2;
        gr = min(max(gr, 0), HH - 1);
        gc = min(max(gc, 0), WW - 1);
        img[r][c] = (tb[(size_t)gr * WW + gc] > 0.5f) ? 255.0f : 0.0f;
    }
    __syncthreads();

    for (int i = tid; i < 324; i += 256) {
        int r = i / 18, c = i % 18;
        int gr = tile_r + r - 1, gc = tile_c + c - 1;
        float mv = 0.0f;
        if (gr >= 0 && gr < HH && gc >= 0 && gc < WW) {
            float gx = -img[r][c]     + img[r][c+2]
                       -2.0f*img[r+1][c] + 2.0f*img[r+1][c+2]
                       -img[r+2][c]   + img[r+2][c+2];
            float gy = -img[r][c] - 2.0f*img[r][c+1] - img[r][c+2]
                       +img[r+2][c] + 2.0f*img[r+2][c+1] + img[r+2][c+2];
            mv = fabsf(gx) + fabsf(gy);
        }
        sm[r][c] = mv;
    }
    __syncthreads();

    const int row = tile_r + ty, col = tile_c + tx;
    float gx = -img[ty+1][tx+1] + img[ty+1][tx+3]
               -2.0f*img[ty+2][tx+1] + 2.0f*img[ty+2][tx+3]
               -img[ty+3][tx+1] + img[ty+3][tx+3];
    float gy = -img[ty+1][tx+1] - 2.0f*img[ty+1][tx+2] - img[ty+1][tx+3]
               +img[ty+3][tx+1] + 2.0f*img[ty+3][tx+2] + img[ty+3][tx+3];
    float m  = sm[ty+1][tx+1];
    float ax = fabsf(gx), ay = fabsf(gy);
    float left = sm[ty+1][tx], right = sm[ty+1][tx+2];
    float up   = sm[ty][tx+1], down  = sm[ty+2][tx+1];
    float ul   = sm[ty][tx],   ur    = sm[ty][tx+2];
    float dl   = sm[ty+2][tx], dr    = sm[ty+2][tx+2];

    bool keep;
    if (ay < ax * TG22f)          keep = (m > left) && (m >= right);
    else if (ay > ax * TG67f)     keep = (m > up)   && (m >= down);
    else if (gx * gy < 0.0f)      keep = (m > ur)   && (m >= dl);
    else                          keep = (m > ul)   && (m >= dr);

    bool interior = (row >= 1) && (row <= HH - 2) && (col >= 1) && (col <= WW - 2);
    bool cand   = keep && (m > LOWT) && interior;
    bool strong = cand && (m > HIGHT);
    flags[(size_t)b * HH * WW + (size_t)row * WW + col] =
        (unsigned char)((cand ? 1 : 0) | (strong ? 2 : 0));
}

// ---------------- Kernel 2: one hysteresis pass (tile-local convergence in LDS) ----------------
__global__ void hyst_kernel(unsigned char* __restrict__ flags) {
    __shared__ unsigned char f[34][34];
    __shared__ int changed;
    const int tx = threadIdx.x, ty = threadIdx.y;
    const int tid = ty * 32 + tx;
    const int tile_r = blockIdx.y * 32, tile_c = blockIdx.x * 32;
    const int b = blockIdx.z;
    unsigned char* fb = flags + (size_t)b * HH * WW;

    for (int i = tid; i < 34 * 34; i += 1024) {
        int r = i / 34, c = i % 34;
        int gr = tile_r + r - 1, gc = tile_c + c - 1;
        f[r][c] = (gr >= 0 && gr < HH && gc >= 0 && gc < WW)
                    ? fb[(size_t)gr * WW + gc] : (unsigned char)0;
    }
    __syncthreads();

    const int mr = ty + 1, mc = tx + 1;
    for (int it = 0; it < 40; ++it) {
        if (tid == 0) changed = 0;
        __syncthreads();
        unsigned char v = f[mr][mc];
        bool promote = false;
        if ((v & 3) == 1) { // weak, not yet strong
            unsigned char nb = (unsigned char)(f[mr-1][mc-1] | f[mr-1][mc] | f[mr-1][mc+1]
                                             | f[mr][mc-1]                 | f[mr][mc+1]
                                             | f[mr+1][mc-1] | f[mr+1][mc] | f[mr+1][mc+1]);
            promote = (nb & 2) != 0;
        }
        __syncthreads();
        if (promote) { f[mr][mc] = (unsigned char)(v | 2); changed = 1; }
        __syncthreads();
        int ch = changed;
        __syncthreads(); // protect read of `changed` from next-iter reset
        if (!ch) break;
    }
    fb[(size_t)(tile_r + ty) * WW + (tile_c + tx)] = f[mr][mc];
}

// ---------------- Kernel 3: TDM-prefetched pred/target + 5x5 dilate + BCE + WMMA reduce ----------
// Dynamic LDS layout (no static LDS in this kernel => dynamic LDS starts at offset 0,
// so TDM lds_addr fields are compile-time constants):
#define BCE_LDS_PRED 0u     // 16*16 f32 = 1024 B
#define BCE_LDS_TARG 1024u  // 16*16 f32 = 1024 B
#define BCE_LDS_FLAG 2048u  // 20*20 u8  =  400 B
#define BCE_LDS_WSUM 2464u  // 8 f32     =   32 B
#define BCE_LDS_SIZE 2496u

__global__ void bce_kernel(const float* __restrict__ pred, const float* __restrict__ target,
                           const unsigned char* __restrict__ flags, float* __restrict__ partials) {
    extern __shared__ char smem[];
    float*         ldsP  = (float*)(smem + BCE_LDS_PRED);
    float*         ldsT  = (float*)(smem + BCE_LDS_TARG);
    unsigned char* st    = (unsigned char*)(smem + BCE_LDS_FLAG);
    float*         wsums = (float*)(smem + BCE_LDS_WSUM);

    const int tx = threadIdx.x, ty = threadIdx.y;
    const int tid = ty * 16 + tx;
    const int tile_r = blockIdx.y * 16, tile_c = blockIdx.x * 16;
    const int b = blockIdx.z;
    const size_t tile_off = (size_t)b * HH * WW + (size_t)tile_r * WW + tile_c;

    // Wave 0 kicks off two async TDM tile loads; the DMA engine fills LDS while
    // the vector pipe loads the flag halo and computes the 5x5 dilation below.
    if (tid < 32) {
        tdm_load_tile_f32((unsigned long long)(pred + tile_off),   BCE_LDS_PRED,
                          16u, 16u, WW, HH, WW);
        tdm_load_tile_f32((unsigned long long)(target + tile_off), BCE_LDS_TARG,
                          16u, 16u, WW, HH, WW);
    }

    const unsigned char* fb = flags + (size_t)b * HH * WW;
    for (int i = tid; i < 400; i += 256) {
        int r = i / 20, c = i % 20;
        int gr = tile_r + r - 2, gc = tile_c + c - 2;
        st[i] = (gr >= 0 && gr < HH && gc >= 0 && gc < WW)
                  ? fb[(size_t)gr * WW + gc] : (unsigned char)0;
    }
    __syncthreads();

    unsigned char any = 0;
    #pragma unroll
    for (int dr = 0; dr < 5; ++dr)
        #pragma unroll
        for (int dc = 0; dc < 5; ++dc)
            any |= st[(ty + dr) * 20 + (tx + dc)];
    float w = (any & 2) ? EDGEW : 1.0f;

    // TDM completion: issuing wave drains TENSORcnt (no-op for the others), then barrier.
    __builtin_amdgcn_s_wait_tensorcnt(0);
    __syncthreads();

    float p = ldsP[tid], t = ldsT[tid];
    float logp   = fmaxf(logf(p), -100.0f);
    float log1mp = fmaxf(log1pf(-p), -100.0f);
    float val = -(t * logp + (1.0f - t) * log1mp) * w;

    float wsum = wave_sum_wmma(val); // all 256 threads active -> EXEC all ones
    int wid = tid >> 5, lane = tid & 31;
    if (lane == 0) wsums[wid] = wsum;
    __syncthreads();
    if (tid == 0) {
        float s = 0.f;
        #pragma unroll
        for (int i = 0; i < 8; ++i) s += wsums[i];
        size_t bid = ((size_t)blockIdx.z * gridDim.y + blockIdx.y) * gridDim.x + blockIdx.x;
        partials[bid] = s;
    }
}

// ---------------- Kernel 4: deterministic final reduction + mean ----------------
__global__ void reduce_kernel(const float* __restrict__ partials, int n,
                              float* __restrict__ out, float inv_n) {
    __shared__ float wsums[8];
    float s = 0.f;
    for (int i = threadIdx.x; i < n; i += 256) s += partials[i];
    float wsum = wave_sum_wmma(s);
    int wid = threadIdx.x >> 5, lane = threadIdx.x & 31;
    if (lane == 0) wsums[wid] = wsum;
    __syncthreads();
    if (threadIdx.x == 0) {
        float t = 0.f;
        #pragma unroll
        for (int i = 0; i < 8; ++i) t += wsums[i];
        out[0] = t * inv_n;
    }
}

extern "C" void kernel_launch(void* const* d_in, const int* in_sizes, int n_in,
                              void* d_out, int out_size, void* d_ws, size_t ws_size,
                              hipStream_t stream) {
    const float* pred   = (const float*)d_in[0];
    const float* target = (const float*)d_in[1];
    const int n = in_sizes[0];              // B*C*H*W
    const int B = n / (HH * WW);            // C == 1

    unsigned char* flags = (unsigned char*)d_ws;
    size_t flags_bytes = (size_t)B * HH * WW;
    float* partials = (float*)((char*)d_ws + ((flags_bytes + 255) & ~(size_t)255));

    dim3 b16(16, 16), g16(WW / 16, HH / 16, B);
    dim3 b32(32, 32), g32(WW / 32, HH / 32, B);

    sobel_nms_kernel<<<g16, b16, 0, stream>>>(target, flags);
    for (int p = 0; p < 3; ++p)
        hyst_kernel<<<g32, b32, 0, stream>>>(flags);
    bce_kernel<<<g16, b16, BCE_LDS_SIZE, stream>>>(pred, target, flags, partials);

    int nblocks = (WW / 16) * (HH / 16) * B;
    reduce_kernel<<<1, 256, 0, stream>>>(partials, nblocks, (float*)d_out, 1.0f / (float)n);
}